// SGConv_JJ_21474836480037
// MI455X (gfx1250) — compile-verified
//
#include <hip/hip_runtime.h>
#include <hip/hip_bf16.h>

// ---------------- problem constants (feature dims are structural) -----------
#define F_IN   128
#define F_OUT  64
#define T_DIM  20
#define L_DIM  40
#define SPLIT_ 15

typedef float v2f __attribute__((ext_vector_type(2)));
typedef float v8f __attribute__((ext_vector_type(8)));

// ---------------- workspace zeroing ----------------------------------------
__global__ void k_zero(float* __restrict__ p, size_t n) {
    size_t i = (size_t)blockIdx.x * blockDim.x + threadIdx.x;
    size_t s = (size_t)gridDim.x * blockDim.x;
    for (; i < n; i += s) p[i] = 0.0f;
}

// ---------------- degree + normalization -----------------------------------
__global__ void k_deg(const int* __restrict__ dst, float* __restrict__ deg, int E) {
    int i = blockIdx.x * blockDim.x + threadIdx.x;
    int s = gridDim.x * blockDim.x;
    for (; i < E; i += s) atomicAdd(&deg[dst[i]], 1.0f);
}

__global__ void k_nrm(float* __restrict__ deg, int N) {
    int i = blockIdx.x * blockDim.x + threadIdx.x;
    if (i < N) deg[i] = rsqrtf(fmaxf(deg[i], 1.0f));
}

// ---------------- edge scatter: agg[dst] += x[src] * nrm[src] ---------------
// one wave (32 lanes) per edge, float4 per lane covers the 128-f32 row
__global__ void k_edge(const float* __restrict__ x, const int* __restrict__ src,
                       const int* __restrict__ dst, const float* __restrict__ nrm,
                       float* __restrict__ agg, int E) {
    int wave   = blockIdx.x * (blockDim.x >> 5) + (threadIdx.x >> 5);
    int lane   = threadIdx.x & 31;
    int nwaves = gridDim.x * (blockDim.x >> 5);
    for (int e = wave; e < E; e += nwaves) {
        int s = src[e], d = dst[e];
        float ns = nrm[s];
        float4 v = ((const float4*)(x + (size_t)s * F_IN))[lane];
        float* ar = agg + (size_t)d * F_IN + lane * 4;
        atomicAdd(ar + 0, v.x * ns);
        atomicAdd(ar + 1, v.y * ns);
        atomicAdd(ar + 2, v.z * ns);
        atomicAdd(ar + 3, v.w * ns);
    }
}

// ---------------- agg[i] *= nrm[i] ------------------------------------------
__global__ void k_scale_rows(float* __restrict__ agg, const float* __restrict__ nrm, int N) {
    size_t total = (size_t)N * F_IN;
    size_t i = (size_t)blockIdx.x * blockDim.x + threadIdx.x;
    size_t s = (size_t)gridDim.x * blockDim.x;
    for (; i < total; i += s) agg[i] *= nrm[i >> 7];
}

// ---------------- pass 1: (time,label) bucket sums + test sums --------------
// stride is a multiple of 128 so f = tid & 127 is loop-invariant
__global__ void k_pass1(const float* __restrict__ agg, const int* __restrict__ times,
                        const int* __restrict__ labels, float* __restrict__ sum_x,
                        float* __restrict__ cnt, float* __restrict__ test_sum,
                        float* __restrict__ n_test, int N) {
    int tid = blockIdx.x * blockDim.x + threadIdx.x;
    int stride = gridDim.x * blockDim.x;         // launch keeps this % 128 == 0
    int f = tid & 127;
    float tacc = 0.0f, tcnt = 0.0f;
    size_t total = (size_t)N * F_IN;
    for (size_t idx = tid; idx < total; idx += stride) {
        int r = (int)(idx >> 7);
        int t = times[r];
        float v = agg[idx];
        if (t < SPLIT_) {
            int l = labels[r];
            atomicAdd(&sum_x[(size_t)(t * L_DIM + l) * F_IN + f], v);
            if (f == 0) atomicAdd(&cnt[t * L_DIM + l], 1.0f);
        } else {
            tacc += v;
            if (f == 0) tcnt += 1.0f;
        }
    }
    atomicAdd(&test_sum[f], tacc);
    if (f == 0) atomicAdd(n_test, tcnt);
}

// ---------------- test mean --------------------------------------------------
__global__ void k_test_mean(const float* __restrict__ test_sum, const float* __restrict__ n_test,
                            float* __restrict__ test_mean) {
    int f = threadIdx.x;
    test_mean[f] = test_sum[f] / fmaxf(1.0f, n_test[0]);
}

// ---------------- per-time stats: train_mean, time_cnt, msq numerator -------
__global__ void k_stats_mid(const float* __restrict__ sum_x, const float* __restrict__ cnt,
                            float* __restrict__ train_mean, float* __restrict__ time_cnt,
                            float* __restrict__ msq) {
    int t = blockIdx.x;          // T_DIM blocks
    int f = threadIdx.x;         // 128 threads
    float tc = 0.0f;
    for (int l = 0; l < L_DIM; ++l) tc += cnt[t * L_DIM + l];
    float tsum = 0.0f;
    for (int l = 0; l < L_DIM; ++l) {
        size_t o = (size_t)(t * L_DIM + l) * F_IN + f;
        float s = sum_x[o];
        tsum += s;
        train_mean[o] = s / fmaxf(1.0f, cnt[t * L_DIM + l]);
    }
    float tmean = tsum / fmaxf(1.0f, tc);
    float p = 0.0f;
    for (int l = 0; l < L_DIM; ++l) {
        size_t o = (size_t)(t * L_DIM + l) * F_IN + f;
        float d = train_mean[o] - tmean;
        p += cnt[t * L_DIM + l] * d * d;
    }
    // reduce p over 128 threads: shuffle within wave, LDS across the 4 waves
    for (int off = 16; off; off >>= 1) p += __shfl_xor(p, off, 32);
    __shared__ float red[4];
    int lane = threadIdx.x & 31, w = threadIdx.x >> 5;
    if (lane == 0) red[w] = p;
    __syncthreads();
    if (threadIdx.x == 0) {
        msq[t] = red[0] + red[1] + red[2] + red[3];   // numerator; /denom later
        time_cnt[t] = tc;
    }
}

// ---------------- pass 2: residual sums (wave per node) ---------------------
__global__ void k_pass2(const float* __restrict__ agg, const int* __restrict__ times,
                        const int* __restrict__ labels, const float* __restrict__ train_mean,
                        const float* __restrict__ test_mean, float* __restrict__ rsq,
                        float* __restrict__ test_var_num, int N) {
    int wave   = blockIdx.x * (blockDim.x >> 5) + (threadIdx.x >> 5);
    int lane   = threadIdx.x & 31;
    int nwaves = gridDim.x * (blockDim.x >> 5);
    for (int r = wave; r < N; r += nwaves) {
        int t = times[r];
        float4 v = ((const float4*)(agg + (size_t)r * F_IN))[lane];
        const float* m = (t < SPLIT_)
            ? train_mean + (size_t)(t * L_DIM + labels[r]) * F_IN
            : test_mean;
        float4 mm = ((const float4*)m)[lane];
        float dx = v.x - mm.x, dy = v.y - mm.y, dz = v.z - mm.z, dw = v.w - mm.w;
        float p = dx * dx + dy * dy + dz * dz + dw * dw;
        for (int off = 16; off; off >>= 1) p += __shfl_xor(p, off, 32);
        if (lane == 0) {
            if (t < SPLIT_) atomicAdd(&rsq[t], p);
            else            atomicAdd(test_var_num, p);
        }
    }
}

// ---------------- alpha ------------------------------------------------------
__global__ void k_alpha(const float* __restrict__ time_cnt, const float* __restrict__ msq,
                        const float* __restrict__ rsq, const float* __restrict__ test_var_num,
                        const float* __restrict__ n_test, float* __restrict__ alpha) {
    int t = threadIdx.x;
    if (t < T_DIM) {
        float denom = fmaxf(1.0f, time_cnt[t] - 1.0f);
        float m = msq[t] / denom;
        float r = rsq[t] / denom;
        float tv = test_var_num[0] / fmaxf(1.0f, n_test[0] - 1.0f);
        float asq = (tv - m) / fmaxf(1e-6f, r);
        alpha[t] = asq > 0.0f ? sqrtf(asq) : 0.0f;
    }
}

// ---------------- pass 3: mixing + column sums for standardization ----------
__global__ void k_pass3(float* __restrict__ agg, const int* __restrict__ times,
                        const int* __restrict__ labels, const float* __restrict__ train_mean,
                        const float* __restrict__ alpha, float* __restrict__ col_sum,
                        float* __restrict__ col_sumsq, int N) {
    int tid = blockIdx.x * blockDim.x + threadIdx.x;
    int stride = gridDim.x * blockDim.x;         // % 128 == 0
    int f = tid & 127;
    float s = 0.0f, ss = 0.0f;
    size_t total = (size_t)N * F_IN;
    for (size_t idx = tid; idx < total; idx += stride) {
        int r = (int)(idx >> 7);
        int t = times[r];
        float v = agg[idx];
        if (t < SPLIT_) {
            float a  = alpha[t];
            float nm = train_mean[(size_t)(t * L_DIM + labels[r]) * F_IN + f];
            v = a * v + (1.0f - a) * nm;
            agg[idx] = v;
        }
        s += v; ss += v * v;
    }
    atomicAdd(&col_sum[f], s);
    atomicAdd(&col_sumsq[f], ss);
}

// ---------------- mu / inv-sd ------------------------------------------------
__global__ void k_musd(const float* __restrict__ col_sum, const float* __restrict__ col_sumsq,
                       float* __restrict__ mu, float* __restrict__ isd, int N) {
    int f = threadIdx.x;
    float m = col_sum[f] / (float)N;
    float var = (col_sumsq[f] - (float)N * m * m) / (float)(N - 1);
    mu[f] = m;
    isd[f] = rsqrtf(fmaxf(var, 1e-20f));
}

// ---------------- fold standardization into W, b ----------------------------
// W'[n,k] = W[n,k]*isd[k];  b'[n] = b[n] - sum_k W'[n,k]*mu[k]
__global__ void k_foldW(const float* __restrict__ W, const float* __restrict__ bias,
                        const float* __restrict__ mu, const float* __restrict__ isd,
                        float* __restrict__ Wp, float* __restrict__ bp) {
    int n = threadIdx.x;   // 64 threads
    float acc = 0.0f;
    for (int k = 0; k < F_IN; ++k) {
        float w = W[n * F_IN + k] * isd[k];
        Wp[n * F_IN + k] = w;
        acc += w * mu[k];
    }
    bp[n] = bias[n] - acc;
}

// ---------------- WMMA GEMM: out[16r x 16c] tiles, K=128 in steps of 4 ------
// V_WMMA_F32_16X16X4_F32 fragment layout (ISA 7.12.2):
//   A (16x4):  lane m = lane&15, h = lane>>4;  a = {A[m,k0+2h], A[m,k0+2h+1]}
//   B (4x16):  lane n = lane&15;               b = {B[k0+2h,n], B[k0+2h+1,n]}
//   C/D:       element (vr + 8h, lane&15) in VGPR vr
__global__ void k_wmma_out(const float* __restrict__ feat, const float* __restrict__ Wp,
                           const float* __restrict__ bp, float* __restrict__ out, int nTiles) {
    int wave = blockIdx.x * (blockDim.x >> 5) + (threadIdx.x >> 5);
    if (wave >= nTiles) return;                 // wave-uniform: EXEC stays all-1s
    int rowTile = wave >> 2;                    // F_OUT/16 = 4 column tiles
    int colTile = wave & 3;
    int lane = threadIdx.x & 31;
    int m = lane & 15;
    int h = lane >> 4;
    const float* arow = feat + (size_t)(rowTile * 16 + m) * F_IN;
    int n = colTile * 16 + m;
    const float* brow = Wp + (size_t)n * F_IN;  // B[k,n] = Wp[n,k] (B = Wp^T)
    float biasv = bp[n];
    v8f c;
    for (int i = 0; i < 8; ++i) c[i] = biasv;
#pragma unroll 4
    for (int k0 = 0; k0 < F_IN; k0 += 4) {
        v2f a, b;
        a.x = arow[k0 + 2 * h];
        a.y = arow[k0 + 2 * h + 1];
        b.x = brow[k0 + 2 * h];
        b.y = brow[k0 + 2 * h + 1];
        c = __builtin_amdgcn_wmma_f32_16x16x4_f32(false, a, false, b,
                                                  (short)0, c, false, false);
    }
    float* obase = out + (size_t)(rowTile * 16) * F_OUT + colTile * 16 + m;
    for (int vr = 0; vr < 8; ++vr)
        obase[(size_t)(vr + 8 * h) * F_OUT] = c[vr];
}

// scalar tail for N % 16 != 0 (not taken for N=100000; kept for generality)
__global__ void k_tail_out(const float* __restrict__ feat, const float* __restrict__ Wp,
                           const float* __restrict__ bp, float* __restrict__ out,
                           int startRow, int N) {
    int idx = blockIdx.x * blockDim.x + threadIdx.x;
    int rem = N - startRow;
    if (idx >= rem * F_OUT) return;
    int r = startRow + idx / F_OUT, n = idx % F_OUT;
    float acc = bp[n];
    for (int k = 0; k < F_IN; ++k) acc += feat[(size_t)r * F_IN + k] * Wp[n * F_IN + k];
    out[(size_t)r * F_OUT + n] = acc;
}

// ---------------- host-side orchestration -----------------------------------
extern "C" void kernel_launch(void* const* d_in, const int* in_sizes, int n_in,
                              void* d_out, int out_size, void* d_ws, size_t ws_size,
                              hipStream_t stream) {
    const float* x      = (const float*)d_in[0];
    const int*   src    = (const int*)  d_in[1];
    const int*   dst    = (const int*)  d_in[2];
    const int*   times  = (const int*)  d_in[3];
    const int*   labels = (const int*)  d_in[4];
    const float* W      = (const float*)d_in[5];
    const float* bias   = (const float*)d_in[6];
    float* out = (float*)d_out;

    const int N = in_sizes[3];
    const int E = in_sizes[1];

    // workspace carve-up (floats)
    float* agg        = (float*)d_ws;                       // N*128
    float* nrm        = agg + (size_t)N * F_IN;             // N
    float* sum_x      = nrm + N;                            // T*L*128
    float* cnt        = sum_x + (size_t)T_DIM * L_DIM * F_IN; // T*L
    float* train_mean = cnt + T_DIM * L_DIM;                // T*L*128
    float* test_sum   = train_mean + (size_t)T_DIM * L_DIM * F_IN; // 128
    float* test_mean  = test_sum + F_IN;                    // 128
    float* time_cnt   = test_mean + F_IN;                   // T
    float* msq        = time_cnt + T_DIM;                   // T
    float* rsq        = msq + T_DIM;                        // T
    float* alpha      = rsq + T_DIM;                        // T
    float* col_sum    = alpha + T_DIM;                      // 128
    float* col_sumsq  = col_sum + F_IN;                     // 128
    float* mu         = col_sumsq + F_IN;                   // 128
    float* isd        = mu + F_IN;                          // 128
    float* scal       = isd + F_IN;                         // 2: n_test, test_var_num
    float* Wp         = scal + 2;                           // 64*128
    float* bp         = Wp + F_OUT * F_IN;                  // 64
    size_t zcount = (size_t)(scal + 2 - (float*)d_ws);      // zero all accumulators

    k_zero<<<2048, 256, 0, stream>>>((float*)d_ws, zcount);
    k_deg<<<2048, 256, 0, stream>>>(dst, nrm, E);
    k_nrm<<<(N + 255) / 256, 256, 0, stream>>>(nrm, N);
    k_edge<<<2048, 256, 0, stream>>>(x, src, dst, nrm, agg, E);
    k_scale_rows<<<2048, 256, 0, stream>>>(agg, nrm, N);
    k_pass1<<<1024, 256, 0, stream>>>(agg, times, labels, sum_x, cnt,
                                      test_sum, scal /*n_test*/, N);
    k_test_mean<<<1, F_IN, 0, stream>>>(test_sum, scal, test_mean);
    k_stats_mid<<<T_DIM, F_IN, 0, stream>>>(sum_x, cnt, train_mean, time_cnt, msq);
    k_pass2<<<2048, 256, 0, stream>>>(agg, times, labels, train_mean, test_mean,
                                      rsq, scal + 1 /*test_var_num*/, N);
    k_alpha<<<1, 32, 0, stream>>>(time_cnt, msq, rsq, scal + 1, scal, alpha);
    k_pass3<<<1024, 256, 0, stream>>>(agg, times, labels, train_mean, alpha,
                                      col_sum, col_sumsq, N);
    k_musd<<<1, F_IN, 0, stream>>>(col_sum, col_sumsq, mu, isd, N);
    k_foldW<<<1, F_OUT, 0, stream>>>(W, bias, mu, isd, Wp, bp);

    int fullRowTiles = N / 16;
    int nTiles = fullRowTiles * 4;              // 4 column tiles of 16 (F_OUT=64)
    if (nTiles > 0)
        k_wmma_out<<<(nTiles + 7) / 8, 256, 0, stream>>>(agg, Wp, bp, out, nTiles);
    int rem = N - fullRowTiles * 16;
    if (rem > 0)
        k_tail_out<<<(rem * F_OUT + 255) / 256, 256, 0, stream>>>(agg, Wp, bp, out,
                                                                  fullRowTiles * 16, N);
}